// GNNRoutingModel_59365037965493
// MI455X (gfx1250) — compile-verified
//
#include <hip/hip_runtime.h>

typedef __attribute__((ext_vector_type(2))) float v2f;
typedef __attribute__((ext_vector_type(8))) float v8f;
typedef int v4i_t __attribute__((vector_size(16)));
typedef __attribute__((address_space(1))) v4i_t* as1_v4i_p;
typedef __attribute__((address_space(3))) v4i_t* as3_v4i_p;

#if __has_builtin(__builtin_amdgcn_global_load_async_to_lds_b128)
#define HAVE_ASYNC_LDS 1
#else
#define HAVE_ASYNC_LDS 0
#endif

__device__ __forceinline__ void wait_async0() {
#if __has_builtin(__builtin_amdgcn_s_wait_asynccnt)
  __builtin_amdgcn_s_wait_asynccnt(0);
#else
  asm volatile("s_wait_asynccnt 0x0" ::: "memory");
#endif
}

// ---------------------------------------------------------------------------
// LDS-tiled batched GEMM via fp32 WMMA:  C[b] = A[b] (MxK) * W^T  (W is NxK)
// 256 threads = 8 waves; macro-tile BM=128 x BN=128, K-chunk BK=32.
// Wave grid 2(M) x 4(N): each wave owns a 64x32 C tile = 8 v8f accumulators,
// issuing 8 v_wmma_f32_16x16x4_f32 per 6 ds_load_b64 fragment reads.
// A/B chunks staged in LDS via gfx1250 async global->LDS b128 copies
// (ASYNCcnt + barrier), falling back to plain b128 load/ds_store.
// ---------------------------------------------------------------------------
#define BM 128
#define BN 128
#define BK 32
#define LSTR (BK + 4)  // 36 floats: rows stay 16B-aligned, banks spread

__global__ void __launch_bounds__(256, 2)
wmma_gemm_f32_lds(const float* __restrict__ A, const float* __restrict__ W,
                  const float* __restrict__ bias, float* __restrict__ C,
                  int M, int N, int K, long strideA, long strideC, int relu)
{
  __shared__ float As[BM * LSTR];
  __shared__ float Bs[BN * LSTR];

  const float* Ab = A + (long)blockIdx.z * strideA;
  float*       Cb = C + (long)blockIdx.z * strideC;

  const int m0 = blockIdx.y * BM;
  const int n0 = blockIdx.x * BN;

  const int tid  = threadIdx.x;
  const int lane = tid & 31;
  const int wv   = tid >> 5;
  const int wm   = (wv >> 2) * 64;   // wave M offset in macro-tile
  const int wn   = (wv & 3) * 32;    // wave N offset in macro-tile
  const int half = lane >> 4;
  const int r    = lane & 15;

  v8f acc[4][2];
#pragma unroll
  for (int i = 0; i < 4; ++i)
#pragma unroll
    for (int j = 0; j < 2; ++j) acc[i][j] = (v8f){};

  for (int k0 = 0; k0 < K; k0 += BK) {
    // -------- cooperative stage of A[128xBK] and W[128xBK] into LDS --------
#pragma unroll
    for (int i = 0; i < 4; ++i) {
      const int slot = tid + i * 256;       // 1024 float4 slots per tile
      const int row  = slot >> 3;
      const int c4   = (slot & 7) << 2;
      const float* ga = Ab + (long)(m0 + row) * K + k0 + c4;
      const float* gb = W  + (long)(n0 + row) * K + k0 + c4;
      float* la = &As[row * LSTR + c4];
      float* lb = &Bs[row * LSTR + c4];
#if HAVE_ASYNC_LDS
      __builtin_amdgcn_global_load_async_to_lds_b128(
          (as1_v4i_p)(v4i_t*)ga, (as3_v4i_p)(v4i_t*)la, 0, 0);
      __builtin_amdgcn_global_load_async_to_lds_b128(
          (as1_v4i_p)(v4i_t*)gb, (as3_v4i_p)(v4i_t*)lb, 0, 0);
#else
      *(float4*)la = *(const float4*)ga;
      *(float4*)lb = *(const float4*)gb;
#endif
    }
#if HAVE_ASYNC_LDS
    wait_async0();
#endif
    __syncthreads();

    // ------------------------- 8 K-steps of WMMA --------------------------
#pragma unroll
    for (int kk = 0; kk < BK; kk += 4) {
      v2f a[4], b[2];
#pragma unroll
      for (int i = 0; i < 4; ++i)
        a[i] = *(const v2f*)&As[(wm + 16 * i + r) * LSTR + kk + 2 * half];
#pragma unroll
      for (int j = 0; j < 2; ++j)
        b[j] = *(const v2f*)&Bs[(wn + 16 * j + r) * LSTR + kk + 2 * half];
#pragma unroll
      for (int i = 0; i < 4; ++i)
#pragma unroll
        for (int j = 0; j < 2; ++j)
          acc[i][j] = __builtin_amdgcn_wmma_f32_16x16x4_f32(
              false, a[i], false, b[j], (short)0, acc[i][j], false, false);
    }
    __syncthreads();
  }

  // ------------------------------ epilogue --------------------------------
  const float bv0 = bias ? bias[n0 + wn + r]      : 0.0f;
  const float bv1 = bias ? bias[n0 + wn + 16 + r] : 0.0f;
#pragma unroll
  for (int i = 0; i < 4; ++i) {
#pragma unroll
    for (int v = 0; v < 8; ++v) {
      const int row = m0 + wm + 16 * i + v + 8 * half;
      float x0 = acc[i][0][v] + bv0;
      float x1 = acc[i][1][v] + bv1;
      if (relu) { x0 = fmaxf(x0, 0.0f); x1 = fmaxf(x1, 0.0f); }
      Cb[(long)row * N + n0 + wn + r]      = x0;
      Cb[(long)row * N + n0 + wn + 16 + r] = x1;
    }
  }
}

// ---------------------------------------------------------------------------
// Direct-from-global WMMA GEMM (fallback for small N, e.g. N=64 gat2 linear).
// Each wave computes a 32x32 C tile.
// ---------------------------------------------------------------------------
__global__ void __launch_bounds__(128)
wmma_gemm_f32(const float* __restrict__ A, const float* __restrict__ W,
              const float* __restrict__ bias, float* __restrict__ C,
              int M, int N, int K, long strideA, long strideC, int relu)
{
  const float* Ab = A + (long)blockIdx.z * strideA;
  float*       Cb = C + (long)blockIdx.z * strideC;

  const int lane = threadIdx.x & 31;
  const int wave = threadIdx.x >> 5;
  const int tiles_x = N >> 5;
  const int wid = blockIdx.x * 4 + wave;
  const int ty = wid / tiles_x;
  const int tx = wid % tiles_x;
  if (ty >= (M >> 5)) return;
  const int m0 = ty << 5, n0 = tx << 5;
  const int half = lane >> 4;
  const int r    = lane & 15;

  v8f c00 = {}, c01 = {}, c10 = {}, c11 = {};
  const float* a0p = Ab + (long)(m0 + r) * K + 2 * half;
  const float* a1p = Ab + (long)(m0 + 16 + r) * K + 2 * half;
  const float* b0p = W  + (long)(n0 + r) * K + 2 * half;
  const float* b1p = W  + (long)(n0 + 16 + r) * K + 2 * half;

  for (int k = 0; k < K; k += 4) {
    v2f a0, a1, b0, b1;
    a0.x = a0p[k]; a0.y = a0p[k + 1];
    a1.x = a1p[k]; a1.y = a1p[k + 1];
    b0.x = b0p[k]; b0.y = b0p[k + 1];
    b1.x = b1p[k]; b1.y = b1p[k + 1];
    c00 = __builtin_amdgcn_wmma_f32_16x16x4_f32(false, a0, false, b0, (short)0, c00, false, false);
    c01 = __builtin_amdgcn_wmma_f32_16x16x4_f32(false, a0, false, b1, (short)0, c01, false, false);
    c10 = __builtin_amdgcn_wmma_f32_16x16x4_f32(false, a1, false, b0, (short)0, c10, false, false);
    c11 = __builtin_amdgcn_wmma_f32_16x16x4_f32(false, a1, false, b1, (short)0, c11, false, false);
  }

  const float bv0 = bias ? bias[n0 + r]      : 0.0f;
  const float bv1 = bias ? bias[n0 + 16 + r] : 0.0f;
#pragma unroll
  for (int v = 0; v < 8; ++v) {
    const int row = m0 + v + 8 * half;
    float x00 = c00[v] + bv0, x01 = c01[v] + bv1;
    float x10 = c10[v] + bv0, x11 = c11[v] + bv1;
    if (relu) {
      x00 = fmaxf(x00, 0.0f); x01 = fmaxf(x01, 0.0f);
      x10 = fmaxf(x10, 0.0f); x11 = fmaxf(x11, 0.0f);
    }
    Cb[(long)row * N + n0 + r]             = x00;
    Cb[(long)row * N + n0 + 16 + r]        = x01;
    Cb[(long)(row + 16) * N + n0 + r]      = x10;
    Cb[(long)(row + 16) * N + n0 + 16 + r] = x11;
  }
}

// ---------------------------------------------------------------------------
// Elementwise / graph kernels
// ---------------------------------------------------------------------------
__global__ void fill_f32(float* p, float v, long n) {
  long i = (long)blockIdx.x * blockDim.x + threadIdx.x;
  if (i < n) p[i] = v;
}
__global__ void fill_u32(unsigned* p, unsigned v, long n) {
  long i = (long)blockIdx.x * blockDim.x + threadIdx.x;
  if (i < n) p[i] = v;
}

// fc_hop fusion: Wsum[n][c] = fc_hop_w[n][c] + fc_hop_w[n][c+HID]
__global__ void wsum_kernel(const float* __restrict__ w, float* __restrict__ ws,
                            int N, int HID) {
  long i = (long)blockIdx.x * blockDim.x + threadIdx.x;
  if (i >= (long)N * HID) return;
  int n = (int)(i / HID), c = (int)(i % HID);
  ws[i] = w[(long)n * 2 * HID + c] + w[(long)n * 2 * HID + HID + c];
}

__global__ void gcn_deg(const int* __restrict__ ei, const float* __restrict__ ew,
                        float* deg, int T, int N, int E) {
  long i = (long)blockIdx.x * blockDim.x + threadIdx.x;
  if (i >= (long)T * E) return;
  int t = (int)(i / E), e = (int)(i % E);
  int dst = ei[((long)t * 2 + 1) * E + e];
  atomicAdd(&deg[(long)t * N + dst], ew[(long)t * E + e]);
}

__global__ void gcn_dinv(const float* deg, float* dinv, long n) {
  long i = (long)blockIdx.x * blockDim.x + threadIdx.x;
  if (i >= n) return;
  float d = deg[i] + 1.0f;  // self-loop weight folded in
  dinv[i] = (d > 0.0f) ? rsqrtf(fmaxf(d, 1e-12f)) : 0.0f;
}

__global__ void gcn_init_out(const float* __restrict__ xw, const float* __restrict__ dinv,
                             const float* __restrict__ bias, float* __restrict__ out,
                             long TN, int C) {
  long i = (long)blockIdx.x * blockDim.x + threadIdx.x;
  if (i >= TN * C) return;
  long node = i / C; int c = (int)(i % C);
  float di = dinv[node];
  out[i] = bias[c] + di * di * xw[i];
}

__global__ void gcn_scatter(const int* __restrict__ ei, const float* __restrict__ ew,
                            const float* __restrict__ dinv, const float* __restrict__ xw,
                            float* __restrict__ out, int T, int N, int E, int C) {
  long i = (long)blockIdx.x * blockDim.x + threadIdx.x;
  if (i >= (long)T * E * C) return;
  int c = (int)(i % C);
  long ec = i / C;
  int e = (int)(ec % E), t = (int)(ec / E);
  int src = ei[((long)t * 2) * E + e];
  int dst = ei[((long)t * 2 + 1) * E + e];
  float w = ew[(long)t * E + e];
  float norm = dinv[(long)t * N + src] * w * dinv[(long)t * N + dst];
  atomicAdd(&out[((long)t * N + dst) * C + c], norm * xw[((long)t * N + src) * C + c]);
}

__global__ void relu_ip(float* p, long n) {
  long i = (long)blockIdx.x * blockDim.x + threadIdx.x;
  if (i < n) p[i] = fmaxf(p[i], 0.0f);
}

// ----------------------------- GAT ----------------------------------------
__global__ void gat_scores(const float* __restrict__ xw, const float* __restrict__ att,
                           float* __restrict__ a, long TN, int H, int Ch) {
  long i = (long)blockIdx.x * blockDim.x + threadIdx.x;
  if (i >= TN * H) return;
  long node = i / H; int h = (int)(i % H);
  const float* xp = xw + (node * H + h) * Ch;
  const float* ap = att + (long)h * Ch;
  float s = 0.0f;
  for (int c = 0; c < Ch; ++c) s += xp[c] * ap[c];
  a[i] = s;
}

__device__ inline void gat_edge(long eidx, const int* ei, int T, int E,
                                int& s, int& d) {
  long EG = (long)T * E;
  if (eidx < EG) {
    int t = (int)(eidx / E), e = (int)(eidx % E);
    s = ei[((long)t * 2) * E + e];
    d = ei[((long)t * 2 + 1) * E + e];
  } else {                       // self loop over all T*N flattened nodes
    s = d = (int)(eidx - EG);
  }
}

__device__ inline unsigned fkey(float f) {
  unsigned u = __float_as_uint(f);
  return (u & 0x80000000u) ? ~u : (u | 0x80000000u);
}
__device__ inline float funkey(unsigned k) {
  return __uint_as_float((k & 0x80000000u) ? (k ^ 0x80000000u) : ~k);
}

__global__ void gat_edge_max(const int* __restrict__ ei, const float* __restrict__ as,
                             const float* __restrict__ ad, unsigned* __restrict__ emax,
                             int T, int E, long totE, int H) {
  long i = (long)blockIdx.x * blockDim.x + threadIdx.x;
  if (i >= totE * H) return;
  long eidx = i / H; int h = (int)(i % H);
  int s, d; gat_edge(eidx, ei, T, E, s, d);
  float e = as[(long)s * H + h] + ad[(long)d * H + h];
  e = (e > 0.0f) ? e : 0.2f * e;                     // leaky_relu(0.2)
  atomicMax(&emax[(long)d * H + h], fkey(e));
}

__global__ void gat_edge_exp(const int* __restrict__ ei, const float* __restrict__ as,
                             const float* __restrict__ ad, const unsigned* __restrict__ emax,
                             float* __restrict__ p, float* __restrict__ den,
                             int T, int E, long totE, int H) {
  long i = (long)blockIdx.x * blockDim.x + threadIdx.x;
  if (i >= totE * H) return;
  long eidx = i / H; int h = (int)(i % H);
  int s, d; gat_edge(eidx, ei, T, E, s, d);
  float e = as[(long)s * H + h] + ad[(long)d * H + h];
  e = (e > 0.0f) ? e : 0.2f * e;
  float pe = __expf(e - funkey(emax[(long)d * H + h]));
  p[eidx * H + h] = pe;
  atomicAdd(&den[(long)d * H + h], pe);
}

__global__ void gat_aggregate(const int* __restrict__ ei, const float* __restrict__ p,
                              const float* __restrict__ den, const float* __restrict__ xw,
                              float* __restrict__ out, int T, int E, long totE,
                              int H, int Ch) {
  long i = (long)blockIdx.x * blockDim.x + threadIdx.x;
  if (i >= totE * (long)H * Ch) return;
  int c = (int)(i % Ch);
  long hh = i / Ch; int h = (int)(hh % H);
  long eidx = hh / H;
  int s, d; gat_edge(eidx, ei, T, E, s, d);
  float alpha = p[eidx * H + h] / (den[(long)d * H + h] + 1e-16f);
  atomicAdd(&out[((long)d * H + h) * Ch + c],
            alpha * xw[((long)s * H + h) * Ch + c]);
}

__global__ void bias_act(float* __restrict__ p, const float* __restrict__ bias,
                         int C, long n, int relu) {
  long i = (long)blockIdx.x * blockDim.x + threadIdx.x;
  if (i >= n) return;
  float v = p[i] + bias[i % C];
  p[i] = relu ? fmaxf(v, 0.0f) : v;
}

__global__ void link_dot(const float* __restrict__ z, const int* __restrict__ eli,
                         float* __restrict__ out, int EL, int C) {
  int e = blockIdx.x * blockDim.x + threadIdx.x;
  if (e >= EL) return;
  int a = eli[e], b = eli[EL + e];
  float s = 0.0f;
  const float* za = z + (long)a * C;
  const float* zb = z + (long)b * C;
#pragma unroll 8
  for (int c = 0; c < C; ++c) s += za[c] * zb[c];
  out[e] = s;
}

// ---------------------------------------------------------------------------
// Host-side orchestration
// ---------------------------------------------------------------------------
static inline long cdiv(long a, long b) { return (a + b - 1) / b; }

extern "C" void kernel_launch(void* const* d_in, const int* in_sizes, int n_in,
                              void* d_out, int out_size, void* d_ws, size_t ws_size,
                              hipStream_t stream) {
  const float* x        = (const float*)d_in[0];
  const int*   ei       = (const int*)d_in[1];
  const float* ew       = (const float*)d_in[2];
  const int*   eli      = (const int*)d_in[3];
  const float* conv1_w  = (const float*)d_in[4];
  const float* conv1_b  = (const float*)d_in[5];
  const float* conv2_w  = (const float*)d_in[6];
  const float* conv2_b  = (const float*)d_in[7];
  const float* gat1_w   = (const float*)d_in[8];
  const float* gat1_as  = (const float*)d_in[9];
  const float* gat1_ad  = (const float*)d_in[10];
  const float* gat1_b   = (const float*)d_in[11];
  const float* gat2_w   = (const float*)d_in[12];
  const float* gat2_as  = (const float*)d_in[13];
  const float* gat2_ad  = (const float*)d_in[14];
  const float* gat2_b   = (const float*)d_in[15];
  const float* fc_w     = (const float*)d_in[16];
  const float* fc_b     = (const float*)d_in[17];

  const int HID   = in_sizes[5];                    // 128
  const int DOUT  = in_sizes[15];                   // 64
  const int N     = in_sizes[17];                   // 2048
  const int DIN   = in_sizes[4] / HID;              // 64
  const int T     = in_sizes[0] / (N * DIN);        // 16
  const int E     = in_sizes[2] / T;                // 16384
  const int EL    = in_sizes[3] / 2;                // 8192
  const int H1    = 4;
  const int G1    = in_sizes[11];                   // 512
  const long TN   = (long)T * N;
  const long totE = (long)T * E + TN;

  float* out_link = (float*)d_out;                  // [EL]
  float* out_hop  = (float*)d_out + EL;             // [T,N,N]

  char* ws = (char*)d_ws;
  size_t off = 0;
  auto alloc = [&](size_t bytes) -> void* {
    off = (off + 255) & ~(size_t)255;
    void* p = ws + off;
    off += bytes;
    return p;
  };

  const int TB = 256;
  auto g1 = [&](long n) { return dim3((unsigned)cdiv(n, TB)); };

  // dispatch GEMM: LDS-tiled WMMA when shapes allow, else direct WMMA
  auto gemm = [&](const float* A, const float* W, const float* bias, float* C,
                  int M, int Nc, int K, long sA, long sC, int batch, int relu) {
    if ((M % BM) == 0 && (Nc % BN) == 0 && (K % BK) == 0) {
      dim3 grid((unsigned)(Nc / BN), (unsigned)(M / BM), (unsigned)batch);
      wmma_gemm_f32_lds<<<grid, 256, 0, stream>>>(A, W, bias, C, M, Nc, K, sA, sC, relu);
    } else {
      long waves = ((long)M >> 5) * (Nc >> 5);
      dim3 grid((unsigned)cdiv(waves, 4), 1, (unsigned)batch);
      wmma_gemm_f32<<<grid, 128, 0, stream>>>(A, W, bias, C, M, Nc, K, sA, sC, relu);
    }
  };

  // ======================= Phase 1: hop-count branch =======================
  float* wsum = (float*)alloc((size_t)N * HID * 4);
  float* deg  = (float*)alloc(TN * 4);
  float* dinv = (float*)alloc(TN * 4);
  float* xw1  = (float*)alloc(TN * HID * 4);
  float* h1   = (float*)alloc(TN * HID * 4);
  float* xw2  = (float*)alloc(TN * HID * 4);
  float* h2   = (float*)alloc(TN * HID * 4);

  wsum_kernel<<<g1((long)N * HID), TB, 0, stream>>>(fc_w, wsum, N, HID);

  fill_f32<<<g1(TN), TB, 0, stream>>>(deg, 0.0f, TN);
  gcn_deg<<<g1((long)T * E), TB, 0, stream>>>(ei, ew, deg, T, N, E);
  gcn_dinv<<<g1(TN), TB, 0, stream>>>(deg, dinv, TN);

  // conv1
  gemm(x, conv1_w, nullptr, xw1, N, HID, DIN, (long)N * DIN, (long)N * HID, T, 0);
  gcn_init_out<<<g1(TN * HID), TB, 0, stream>>>(xw1, dinv, conv1_b, h1, TN, HID);
  gcn_scatter<<<g1((long)T * E * HID), TB, 0, stream>>>(ei, ew, dinv, xw1, h1, T, N, E, HID);
  relu_ip<<<g1(TN * HID), TB, 0, stream>>>(h1, TN * HID);

  // conv2
  gemm(h1, conv2_w, nullptr, xw2, N, HID, HID, (long)N * HID, (long)N * HID, T, 0);
  gcn_init_out<<<g1(TN * HID), TB, 0, stream>>>(xw2, dinv, conv2_b, h2, TN, HID);
  gcn_scatter<<<g1((long)T * E * HID), TB, 0, stream>>>(ei, ew, dinv, xw2, h2, T, N, E, HID);
  relu_ip<<<g1(TN * HID), TB, 0, stream>>>(h2, TN * HID);

  // hop_out[t] = h2_t @ (W_lo + W_hi)^T + fc_b   ([h,h] concat fused)
  gemm(h2, wsum, fc_b, out_hop, N, N, HID, (long)N * HID, (long)N * N, T, 0);

  // ======================= Phase 2: link-pred (GAT) ========================
  off = 0;  // hop-phase scratch dead; stream ordering makes reuse safe
  float*    xwg1  = (float*)alloc(TN * G1 * 4);
  float*    z1    = (float*)alloc(TN * G1 * 4);
  float*    as1   = (float*)alloc(TN * H1 * 4);
  float*    ad1   = (float*)alloc(TN * H1 * 4);
  unsigned* emax1 = (unsigned*)alloc(TN * H1 * 4);
  float*    den1  = (float*)alloc(TN * H1 * 4);
  float*    p1    = (float*)alloc(totE * H1 * 4);
  float*    xwg2  = (float*)alloc(TN * DOUT * 4);
  float*    z2    = (float*)alloc(TN * DOUT * 4);
  float*    as2   = (float*)alloc(TN * 4);
  float*    ad2   = (float*)alloc(TN * 4);
  unsigned* emax2 = (unsigned*)alloc(TN * 4);
  float*    den2  = (float*)alloc(TN * 4);
  float*    p2    = (float*)alloc(totE * 4);

  // --- gat1: x(TN x DIN) -> 4 heads x HID, concat ---
  gemm(x, gat1_w, nullptr, xwg1, (int)TN, G1, DIN, 0, 0, 1, 0);
  gat_scores<<<g1(TN * H1), TB, 0, stream>>>(xwg1, gat1_as, as1, TN, H1, HID);
  gat_scores<<<g1(TN * H1), TB, 0, stream>>>(xwg1, gat1_ad, ad1, TN, H1, HID);
  fill_u32<<<g1(TN * H1), TB, 0, stream>>>(emax1, 0u, TN * H1);
  gat_edge_max<<<g1(totE * H1), TB, 0, stream>>>(ei, as1, ad1, emax1, T, E, totE, H1);
  fill_f32<<<g1(TN * H1), TB, 0, stream>>>(den1, 0.0f, TN * H1);
  gat_edge_exp<<<g1(totE * H1), TB, 0, stream>>>(ei, as1, ad1, emax1, p1, den1, T, E, totE, H1);
  fill_f32<<<g1(TN * G1), TB, 0, stream>>>(z1, 0.0f, TN * G1);
  gat_aggregate<<<g1(totE * H1 * HID), TB, 0, stream>>>(ei, p1, den1, xwg1, z1, T, E, totE, H1, HID);
  bias_act<<<g1(TN * G1), TB, 0, stream>>>(z1, gat1_b, G1, TN * G1, 1);

  // --- gat2: z1(TN x 512) -> 1 head x DOUT ---
  gemm(z1, gat2_w, nullptr, xwg2, (int)TN, DOUT, G1, 0, 0, 1, 0);
  gat_scores<<<g1(TN), TB, 0, stream>>>(xwg2, gat2_as, as2, TN, 1, DOUT);
  gat_scores<<<g1(TN), TB, 0, stream>>>(xwg2, gat2_ad, ad2, TN, 1, DOUT);
  fill_u32<<<g1(TN), TB, 0, stream>>>(emax2, 0u, TN);
  gat_edge_max<<<g1(totE), TB, 0, stream>>>(ei, as2, ad2, emax2, T, E, totE, 1);
  fill_f32<<<g1(TN), TB, 0, stream>>>(den2, 0.0f, TN);
  gat_edge_exp<<<g1(totE), TB, 0, stream>>>(ei, as2, ad2, emax2, p2, den2, T, E, totE, 1);
  fill_f32<<<g1(TN * DOUT), TB, 0, stream>>>(z2, 0.0f, TN * DOUT);
  gat_aggregate<<<g1(totE * DOUT), TB, 0, stream>>>(ei, p2, den2, xwg2, z2, T, E, totE, 1, DOUT);
  bias_act<<<g1(TN * DOUT), TB, 0, stream>>>(z2, gat2_b, DOUT, TN * DOUT, 0);

  link_dot<<<g1(EL), TB, 0, stream>>>(z2, eli, out_link, EL, DOUT);
}